// UHGIntrustionDetection_12524124635376
// MI455X (gfx1250) — compile-verified
//
#include <hip/hip_runtime.h>
#include <math.h>

typedef float v2f __attribute__((ext_vector_type(2)));
typedef float v8f __attribute__((ext_vector_type(8)));

#define D_IN   64
#define D_H    128
#define D_OUT  10
#define K0P    68    // (64+1) padded to mult of 4
#define K1P    132   // (128+1) padded to mult of 4
#define EPS_F  1e-8f

// ---------------- utility kernels ----------------

__global__ void k_zero(float4* __restrict__ p, int n4) {
    int i = blockIdx.x * blockDim.x + threadIdx.x;
    if (i < n4) p[i] = make_float4(0.f, 0.f, 0.f, 0.f);
}

// haug0[n][0..63]=x[n], [64]=1, [65..67]=0   (stride 68)
__global__ void k_build_haug(const float* __restrict__ x, float* __restrict__ h, int N) {
    int i = blockIdx.x * blockDim.x + threadIdx.x;
    if (i >= N * K0P) return;
    int n = i / K0P, c = i % K0P;
    float v = (c < D_IN) ? x[n * D_IN + c] : (c == D_IN ? 1.0f : 0.0f);
    h[i] = v;
}

// zero-padded weight copy: (sr x sc) -> (dr x dc)
__global__ void k_pad2d(const float* __restrict__ s, float* __restrict__ d,
                        int sr, int sc, int dr, int dc) {
    int i = blockIdx.x * blockDim.x + threadIdx.x;
    if (i >= dr * dc) return;
    int r = i / dc, c = i % dc;
    d[i] = (r < sr && c < sc) ? s[r * sc + c] : 0.0f;
}

// ---------------- irregular part: gather + atomic scatter-add ----------------
// thread = (edge, float4-chunk); consecutive threads cover consecutive chunks -> coalesced
__global__ void k_scatter(const float4* __restrict__ h, const int* __restrict__ src,
                          const int* __restrict__ dst, float* __restrict__ msg,
                          int E, int stride4) {
    int i = blockIdx.x * blockDim.x + threadIdx.x;
    if (i >= E * stride4) return;
    int e = i / stride4, c = i % stride4;
    int s = src[e], d = dst[e];
    float4 v = h[(size_t)s * stride4 + c];
    float* o = msg + ((size_t)d * stride4 + c) * 4;
    atomicAdd(o + 0, v.x);
    atomicAdd(o + 1, v.y);
    atomicAdd(o + 2, v.z);
    atomicAdd(o + 3, v.w);
}

// mean: divide each row by max(deg,1); deg = aggregated ones-column. wave per node.
__global__ void k_finalize_mean(float* __restrict__ msg, int N, int stride, int degcol) {
    int wave = (blockIdx.x * blockDim.x + threadIdx.x) >> 5;
    int lane = threadIdx.x & 31;
    if (wave >= N) return;
    float deg = msg[(size_t)wave * stride + degcol];
    float s = 1.0f / fmaxf(deg, 1.0f);
    int chunks = stride >> 2;
    float4* row = (float4*)(msg + (size_t)wave * stride);
    for (int c = lane; c < chunks; c += 32) {
        float4 v = row[c];
        v.x *= s; v.y *= s; v.z *= s; v.w *= s;
        row[c] = v;
    }
}

// ---------------- dense part: f32 WMMA GEMM ----------------
// out[16nt : 16nt+16, 16ot : 16ot+16] = A1*W1 + A2*W2 over padded K.
// One wave per 16x16 output tile; V_WMMA_F32_16X16X4_F32 K-loop.
__global__ void k_gemm_wmma(const float* __restrict__ A1, const float* __restrict__ W1,
                            const float* __restrict__ A2, const float* __restrict__ W2,
                            float* __restrict__ out,
                            int node_tiles, int out_tiles, int K,
                            int lda, int ldw, int ldo) {
    int wave = threadIdx.x >> 5;
    int lane = threadIdx.x & 31;
    int t = blockIdx.x * (blockDim.x >> 5) + wave;
    if (t >= node_tiles * out_tiles) return;       // wave-uniform: EXEC stays all-1s
    int nt = t / out_tiles, ot = t % out_tiles;
    int m    = lane & 15;
    int row  = nt * 16 + m;                        // A: M = lane%16
    int col  = ot * 16 + m;                        // B: N = lane%16
    int koff = (lane >> 4) << 1;                   // half-wave K offset: 0 or 2

    v8f acc = {};
    const float* arow1 = A1 + (size_t)row * lda + koff;
    const float* arow2 = A2 + (size_t)row * lda + koff;

    for (int k = 0; k < K; k += 4) {
        v2f a = *(const v2f*)(arow1 + k);          // K=k+koff, k+koff+1
        v2f b;
        b.x = W1[(size_t)(k + koff)     * ldw + col];
        b.y = W1[(size_t)(k + koff + 1) * ldw + col];
        acc = __builtin_amdgcn_wmma_f32_16x16x4_f32(false, a, false, b,
                                                    (short)0, acc, false, false);
    }
    for (int k = 0; k < K; k += 4) {
        v2f a = *(const v2f*)(arow2 + k);
        v2f b;
        b.x = W2[(size_t)(k + koff)     * ldw + col];
        b.y = W2[(size_t)(k + koff + 1) * ldw + col];
        acc = __builtin_amdgcn_wmma_f32_16x16x4_f32(false, a, false, b,
                                                    (short)0, acc, false, false);
    }

    // C/D layout: VGPR r, lanes 0-15 -> M=r, lanes 16-31 -> M=r+8; N = lane%16
    int rbase = nt * 16 + ((lane >> 4) << 3);
    for (int r = 0; r < 8; ++r)
        out[(size_t)(rbase + r) * ldo + col] = acc[r];
}

// ---------------- epilogues ----------------

// layer0: h1 = relu(raw / (||raw||+eps)); append ones col + zero pad (stride 132). wave per node.
__global__ void k_norm_relu_aug(const float* __restrict__ raw, float* __restrict__ h1, int N) {
    int wave = (blockIdx.x * blockDim.x + threadIdx.x) >> 5;
    int lane = threadIdx.x & 31;
    if (wave >= N) return;
    const float4* r = (const float4*)(raw + (size_t)wave * D_H);
    float4 v = r[lane];                                  // 128/32 = 1 float4 per lane
    float s = v.x * v.x + v.y * v.y + v.z * v.z + v.w * v.w;
    for (int off = 16; off > 0; off >>= 1) s += __shfl_xor(s, off, 32);
    float inv = 1.0f / (sqrtf(s) + EPS_F);
    v.x = fmaxf(v.x * inv, 0.f);
    v.y = fmaxf(v.y * inv, 0.f);
    v.z = fmaxf(v.z * inv, 0.f);
    v.w = fmaxf(v.w * inv, 0.f);
    float4* o = (float4*)(h1 + (size_t)wave * K1P);
    o[lane] = v;
    if (lane == 0) o[32] = make_float4(1.f, 0.f, 0.f, 0.f);  // cols 128..131
}

// layer1: normalize 10 valid cols (raw stride 16), then log_softmax -> d_out (N x 10)
__global__ void k_final(const float* __restrict__ raw, float* __restrict__ out, int N) {
    int n = blockIdx.x * blockDim.x + threadIdx.x;
    if (n >= N) return;
    float v[D_OUT];
    float ss = 0.f;
    for (int j = 0; j < D_OUT; ++j) { v[j] = raw[n * 16 + j]; ss += v[j] * v[j]; }
    float inv = 1.0f / (sqrtf(ss) + EPS_F);
    float m = -1e30f;
    for (int j = 0; j < D_OUT; ++j) { v[j] *= inv; m = fmaxf(m, v[j]); }
    float se = 0.f;
    for (int j = 0; j < D_OUT; ++j) se += expf(v[j] - m);
    float lse = logf(se) + m;
    for (int j = 0; j < D_OUT; ++j) out[n * D_OUT + j] = v[j] - lse;
}

// ---------------- launch ----------------

static inline int cdiv(int a, int b) { return (a + b - 1) / b; }

extern "C" void kernel_launch(void* const* d_in, const int* in_sizes, int n_in,
                              void* d_out, int out_size, void* d_ws, size_t ws_size,
                              hipStream_t stream) {
    const float* x   = (const float*)d_in[0];
    const int*   ei  = (const int*)  d_in[1];
    const float* Ws0 = (const float*)d_in[2];
    const float* Wn0 = (const float*)d_in[3];
    const float* Ws1 = (const float*)d_in[4];
    const float* Wn1 = (const float*)d_in[5];
    float* outp = (float*)d_out;

    const int N = in_sizes[0] / D_IN;   // 50000 (divisible by 16)
    const int E = in_sizes[1] / 2;      // 800000
    const int* src = ei;
    const int* dst = ei + E;

    // workspace layout (floats), regions reused across layers
    float* ws   = (float*)d_ws;
    float* bufA = ws;                              // N*132 : haug0 (stride 68) / h1aug (stride 132)
    float* bufB = bufA + (size_t)N * K1P;          // N*132 : msg accumulators (re-zeroed per layer)
    float* bufC = bufB + (size_t)N * K1P;          // N*128 : raw GEMM outputs (layer1 uses N*16)
    float* wS0  = bufC + (size_t)N * D_H;          // 68*128
    float* wN0  = wS0 + K0P * D_H;                 // 68*128
    float* wS1  = wN0 + K0P * D_H;                 // 132*16
    float* wN1  = wS1 + K1P * 16;                  // 132*16

    const int nt = N / 16;                         // 3125 node tiles
    const int n4 = N * K1P / 4;

    // ---- layer 0 ----
    k_zero<<<cdiv(n4, 256), 256, 0, stream>>>((float4*)bufB, n4);
    k_build_haug<<<cdiv(N * K0P, 256), 256, 0, stream>>>(x, bufA, N);
    k_pad2d<<<cdiv(K0P * D_H, 256), 256, 0, stream>>>(Ws0, wS0, D_IN + 1, D_H, K0P, D_H);
    k_pad2d<<<cdiv(K0P * D_H, 256), 256, 0, stream>>>(Wn0, wN0, D_IN + 1, D_H, K0P, D_H);
    k_pad2d<<<cdiv(K1P * 16, 256), 256, 0, stream>>>(Ws1, wS1, D_H + 1, D_OUT, K1P, 16);
    k_pad2d<<<cdiv(K1P * 16, 256), 256, 0, stream>>>(Wn1, wN1, D_H + 1, D_OUT, K1P, 16);

    k_scatter<<<cdiv(E * (K0P / 4), 256), 256, 0, stream>>>(
        (const float4*)bufA, src, dst, bufB, E, K0P / 4);
    k_finalize_mean<<<cdiv(N * 32, 256), 256, 0, stream>>>(bufB, N, K0P, D_IN);

    k_gemm_wmma<<<cdiv(nt * (D_H / 16), 8), 256, 0, stream>>>(
        bufA, wS0, bufB, wN0, bufC, nt, D_H / 16, K0P, K0P, D_H, D_H);
    k_norm_relu_aug<<<cdiv(N * 32, 256), 256, 0, stream>>>(bufC, bufA, N);

    // ---- layer 1 ----
    k_zero<<<cdiv(n4, 256), 256, 0, stream>>>((float4*)bufB, n4);
    k_scatter<<<cdiv(E * (K1P / 4), 256), 256, 0, stream>>>(
        (const float4*)bufA, src, dst, bufB, E, K1P / 4);
    k_finalize_mean<<<cdiv(N * 32, 256), 256, 0, stream>>>(bufB, N, K1P, D_H);

    k_gemm_wmma<<<cdiv(nt, 8), 256, 0, stream>>>(
        bufA, wS1, bufB, wN1, bufC, nt, 1, K1P, K1P, 16, 16);
    k_final<<<cdiv(N, 256), 256, 0, stream>>>(bufC, outp, N);
}